// GCN_24257975287854
// MI455X (gfx1250) — compile-verified
//
#include <hip/hip_runtime.h>
#include <math.h>

typedef float v2f __attribute__((ext_vector_type(2)));
typedef float v8f __attribute__((ext_vector_type(8)));

#define NN 50000
#define NE 800000

// ---------------- degree / norm precompute ----------------

__global__ void k_init_deg(float* deg, int n) {
    int i = blockIdx.x * blockDim.x + threadIdx.x;
    if (i < n) deg[i] = 1.0f;   // self loop contributes 1
}

__global__ void k_count_deg(const int* __restrict__ col, float* deg, int e) {
    int i = blockIdx.x * blockDim.x + threadIdx.x;
    if (i < e) atomicAdd(&deg[col[i]], 1.0f);
}

__global__ void k_dinv(const float* __restrict__ deg, float* dinv, int n) {
    int i = blockIdx.x * blockDim.x + threadIdx.x;
    if (i < n) dinv[i] = rsqrtf(deg[i]);
}

__global__ void k_norm(const int* __restrict__ row, const int* __restrict__ col,
                       const float* __restrict__ dinv, float* norm, int e) {
    int i = blockIdx.x * blockDim.x + threadIdx.x;
    if (i < e) norm[i] = dinv[row[i]] * dinv[col[i]];
}

// ---------------- fp32 WMMA GEMM: H[n x FOUT] = X[n x FIN] @ W[FIN x FOUT] ----------------
// One wave computes a full 16 x FOUT strip: NT = FOUT/16 independent accumulator
// chains (hides XDL latency), A fragment loaded once per K-step and reused for
// all N-tiles. Bounds for FOUT=40 handled branch-free via clamped column index
// + 0/1 multiplicative mask, so the hot loop has uniform control flow (EXEC
// stays all-ones; no saveexec around the WMMAs).
// blockDim.x = 128 (4 waves); wave w -> M-tile (blockIdx.x*4 + w).

template <int FIN, int FOUT>
__global__ void k_gemm_wmma_f32(const float* __restrict__ X, const float* __restrict__ W,
                                float* __restrict__ H, int n) {
    constexpr int NT = (FOUT + 15) / 16;

    const int lane = threadIdx.x & 31;
    const int wave = threadIdx.x >> 5;
    const int half = lane >> 4;        // 0: lanes 0-15, 1: lanes 16-31
    const int l    = lane & 15;

    const int m0 = (blockIdx.x * 4 + wave) * 16;
    if (m0 >= n) return;               // wave-uniform

    v8f acc[NT];
    float bmask[NT];
    int   bcol[NT];
#pragma unroll
    for (int t = 0; t < NT; ++t) {
        acc[t] = (v8f){};
        int c = t * 16 + l;
        bmask[t] = (c < FOUT) ? 1.0f : 0.0f;
        bcol[t]  = (c < FOUT) ? c : (FOUT - 1);
    }

    // A 16x4 fp32 per WMMA: lane l (half h) holds A[M=l][K = k+2h], A[l][k+2h+1]
    const float* xrow = X + (size_t)(m0 + l) * FIN + 2 * half;

    for (int k = 0; k < FIN; k += 4) {
        v2f a;
        a.x = xrow[k];
        a.y = xrow[k + 1];
        const float* w0 = W + (size_t)(k + 2 * half) * FOUT;
#pragma unroll
        for (int t = 0; t < NT; ++t) {
            // B 4x16 fp32: lane l (half h) holds W[K=k+2h][col], W[K=k+2h+1][col]
            v2f b;
            b.x = w0[bcol[t]] * bmask[t];
            b.y = w0[FOUT + bcol[t]] * bmask[t];
            acc[t] = __builtin_amdgcn_wmma_f32_16x16x4_f32(
                /*neg_a=*/false, a, /*neg_b=*/false, b,
                /*c_mod=*/(short)0, acc[t], /*reuse_a=*/false, /*reuse_b=*/false);
        }
    }

    // D layout: VGPR r -> M = r + 8*half, N = l (within tile t)
#pragma unroll
    for (int t = 0; t < NT; ++t) {
        int c = t * 16 + l;
        if (c < FOUT) {
#pragma unroll
            for (int r = 0; r < 8; ++r)
                H[(size_t)(m0 + r + 8 * half) * FOUT + c] = acc[t][r];
        }
    }
}

// ---------------- self-loop init: agg = dinv^2 * h (float4 vectorized) ----------------

__global__ void k_selfinit4(const float4* __restrict__ h, const float* __restrict__ dinv,
                            float4* __restrict__ agg, int n, int fc /* f/4 */) {
    int idx = blockIdx.x * blockDim.x + threadIdx.x;
    if (idx >= n * fc) return;
    int i = idx / fc;
    float d = dinv[i];
    float s = d * d;
    float4 v = h[idx];
    float4 o; o.x = v.x * s; o.y = v.y * s; o.z = v.z * s; o.w = v.w * s;
    agg[idx] = o;
}

// ---------------- edge scatter: agg[col] += norm * h[row] ----------------
// One thread per (edge, 4-feature chunk). Gathers hit L2 (h fits in 192MB L2);
// accumulation via native f32 global atomics.

__global__ void k_scatter(const int* __restrict__ row, const int* __restrict__ col,
                          const float* __restrict__ norm, const float* __restrict__ h,
                          float* agg, int e, int f, int fc) {
    int idx = blockIdx.x * blockDim.x + threadIdx.x;
    if (idx >= e * fc) return;
    int ei = idx / fc;
    int c  = (idx - ei * fc) * 4;
    int r = row[ei];
    int t = col[ei];
    float nv = norm[ei];
    float4 v = *(const float4*)(h + (size_t)r * f + c);
    float* o = agg + (size_t)t * f + c;
    atomicAdd(o + 0, v.x * nv);
    atomicAdd(o + 1, v.y * nv);
    atomicAdd(o + 2, v.z * nv);
    atomicAdd(o + 3, v.w * nv);
}

// ---------------- bias + relu (in place) ----------------

__global__ void k_bias_relu(float* x, const float* __restrict__ b, int n, int f) {
    int idx = blockIdx.x * blockDim.x + threadIdx.x;
    if (idx >= n * f) return;
    int c = idx % f;
    float v = x[idx] + b[c];
    x[idx] = v > 0.0f ? v : 0.0f;
}

// ---------------- final: bias + log_softmax over f columns (in place) ----------------

__global__ void k_bias_logsoftmax(float* out, const float* __restrict__ b, int n, int f) {
    int i = blockIdx.x * blockDim.x + threadIdx.x;
    if (i >= n) return;
    float* r = out + (size_t)i * f;
    float mx = -INFINITY;
    for (int c = 0; c < f; ++c) {
        float v = r[c] + b[c];
        r[c] = v;
        if (v > mx) mx = v;
    }
    float s = 0.0f;
    for (int c = 0; c < f; ++c) s += expf(r[c] - mx);
    float lse = mx + logf(s);
    for (int c = 0; c < f; ++c) r[c] -= lse;
}

// ---------------- orchestration ----------------

extern "C" void kernel_launch(void* const* d_in, const int* in_sizes, int n_in,
                              void* d_out, int out_size, void* d_ws, size_t ws_size,
                              hipStream_t stream) {
    const float* x  = (const float*)d_in[0];        // [NN,128]
    const int* ei   = (const int*)d_in[1];          // [2,NE]
    const float* W1 = (const float*)d_in[2];        // [128,64]
    const float* b1 = (const float*)d_in[3];
    const float* W2 = (const float*)d_in[4];        // [64,64]
    const float* b2 = (const float*)d_in[5];
    const float* W3 = (const float*)d_in[6];        // [64,40]
    const float* b3 = (const float*)d_in[7];
    float* out = (float*)d_out;                     // [NN,40]

    const int* row = ei;            // sources
    const int* col = ei + NE;       // targets

    // workspace layout (floats)
    float* deg  = (float*)d_ws;
    float* dinv = deg  + NN;
    float* norm = dinv + NN;
    float* hA   = norm + NE;               // NN x 64 (also holds NN x 40 for layer 3)
    float* hB   = hA   + (size_t)NN * 64;  // NN x 64

    const int T = 256;

    // degrees + symmetric norm (computed once, reused by all 3 layers)
    k_init_deg  <<<(NN + T - 1) / T, T, 0, stream>>>(deg, NN);
    k_count_deg <<<(NE + T - 1) / T, T, 0, stream>>>(col, deg, NE);
    k_dinv      <<<(NN + T - 1) / T, T, 0, stream>>>(deg, dinv, NN);
    k_norm      <<<(NE + T - 1) / T, T, 0, stream>>>(row, col, dinv, norm, NE);

    const int mTiles = NN / 16;                 // 3125 exactly
    const dim3 gemmGrid((mTiles + 3) / 4);      // 4 waves per block, 1 M-tile each
    const dim3 gemmBlk(128);

    // ---- layer 1: x[NN,128] @ W1 -> hA[NN,64]; aggregate -> hB; relu ----
    {
        k_gemm_wmma_f32<128, 64><<<gemmGrid, gemmBlk, 0, stream>>>(x, W1, hA, NN);
        int fc = 16;
        k_selfinit4<<<(NN * fc + T - 1) / T, T, 0, stream>>>((const float4*)hA, dinv, (float4*)hB, NN, fc);
        k_scatter  <<<(NE * fc + T - 1) / T, T, 0, stream>>>(row, col, norm, hA, hB, NE, 64, fc);
        k_bias_relu<<<(NN * 64 + T - 1) / T, T, 0, stream>>>(hB, b1, NN, 64);
    }

    // ---- layer 2: hB[NN,64] @ W2 -> hA[NN,64]; aggregate -> hB; relu ----
    {
        k_gemm_wmma_f32<64, 64><<<gemmGrid, gemmBlk, 0, stream>>>(hB, W2, hA, NN);
        int fc = 16;
        k_selfinit4<<<(NN * fc + T - 1) / T, T, 0, stream>>>((const float4*)hA, dinv, (float4*)hB, NN, fc);
        k_scatter  <<<(NE * fc + T - 1) / T, T, 0, stream>>>(row, col, norm, hA, hB, NE, 64, fc);
        k_bias_relu<<<(NN * 64 + T - 1) / T, T, 0, stream>>>(hB, b2, NN, 64);
    }

    // ---- layer 3: hB[NN,64] @ W3 -> hA[NN,40]; aggregate -> d_out; bias+log_softmax ----
    {
        k_gemm_wmma_f32<64, 40><<<gemmGrid, gemmBlk, 0, stream>>>(hB, W3, hA, NN);
        int fc = 10;                                    // 40/4
        k_selfinit4<<<(NN * fc + T - 1) / T, T, 0, stream>>>((const float4*)hA, dinv, (float4*)out, NN, fc);
        k_scatter  <<<(NE * fc + T - 1) / T, T, 0, stream>>>(row, col, norm, hA, out, NE, 40, fc);
        k_bias_logsoftmax<<<(NN + T - 1) / T, T, 0, stream>>>(out, b3, NN, 40);
    }
}